// BackProjector_67104569033124
// MI455X (gfx1250) — compile-verified
//
#include <hip/hip_runtime.h>

// ---- Problem constants (from the reference) ----
#define PFD     2.0
#define DIMX    194              // ORI + PF = 192 + 2
#define DIMY    387              // 2*DIMX - 1
#define DIMZ    387
#define MAX_R2d 36864.0          // (ORI/2 * PF)^2 = 192^2
#define NIMG    64
#define HH      192
#define WW      97
#define PLANE   (DIMY * DIMX)              // 75078
#define TOT     ((long)DIMZ * DIMY * DIMX) // 29,055,186 voxels

// -------------------------------------------------------------------------
// Main back-projection scatter. One thread per (h,w) pixel; blockIdx.y = n.
// Output: out[0 .. 2*TOT)    = complex data volume, interleaved (re,im) f64
//         out[2*TOT .. 3*TOT) = weight volume f64
// (Defined first so the disasm snippet shows this kernel.)
// -------------------------------------------------------------------------
__global__ __launch_bounds__(256) void bp_scatter(
    const double* __restrict__ f2d_real,
    const double* __restrict__ f2d_imag,
    const double* __restrict__ Mweight,
    const double* __restrict__ cst,
    double* __restrict__ out)
{
    __shared__ double sc[9];
    const int n = blockIdx.y;

    // CDNA5 async global->LDS stage of the 9 per-image constants
    // (GLOBAL_LOAD_ASYNC_TO_LDS_B64, tracked by ASYNCcnt; ISA 15.18.3 op 97).
    if (threadIdx.x < 9) {
        unsigned ldsOff = (unsigned)(size_t)(&sc[threadIdx.x]);      // low 32b of LDS flat addr
        unsigned gOff   = (unsigned)(((unsigned)n * 9u + threadIdx.x) * 8u);
        asm volatile("global_load_async_to_lds_b64 %0, %1, %2"
                     :: "v"(ldsOff), "v"(gOff), "s"(cst)
                     : "memory");
        asm volatile("s_wait_asynccnt 0x0" ::: "memory");
    }
    __syncthreads();

    const int pix = blockIdx.x * blockDim.x + threadIdx.x;
    if (pix >= HH * WW) return;
    const int h = pix / WW;
    const int w = pix - h * WW;

    const double y = (h < WW) ? (double)h : (double)(h - HH);
    const double x = (double)w;

    const double r00 = sc[0], r01 = sc[1];
    const double r10 = sc[2], r11 = sc[3];
    const double r20 = sc[4], r21 = sc[5];
    const double axx = sc[6], axy = sc[7], ayy = sc[8];

    // Row extents (reference: discr, q0, q1, first_x, last_x)
    const double y2    = y * y;
    const double discr = axy * axy * y2 - axx * (ayy * y2 - MAX_R2d);
    if (!(discr >= 0.0)) return;
    const double q0 = sqrt(discr) / axx;
    const double q1 = -axy * y / axx;
    double firstx = ceil(q1 - q0);
    if (firstx < 0.0) firstx = 0.0;
    if (h >= WW && firstx < 1.0) firstx = 1.0;   // negative-y rows: first_x >= 1
    double lastx = floor(q1 + q0);
    if (lastx > (double)(WW - 1)) lastx = (double)(WW - 1);
    if (x < firstx || x > lastx) return;

    const long gidx = (long)n * (HH * WW) + pix;
    const double Mw = Mweight[gidx];
    if (!(Mw > 0.0)) return;

    // p = Arot * [y, x]
    double pz = r00 * y + r01 * x;
    double py = r10 * y + r11 * x;
    double px = r20 * y + r21 * x;
    const double r2 = pz * pz + py * py + px * px;
    if (r2 > MAX_R2d) return;

    double re = f2d_real[gidx];
    double im = f2d_imag[gidx];
    if (px < 0.0) { pz = -pz; py = -py; px = -px; im = -im; }   // Friedel conjugate

    const double fz0 = floor(pz), fy0 = floor(py), fx0 = floor(px);
    const double fz = pz - fz0, fy = py - fy0, fx = px - fx0;
    const int iz = (int)fz0 + (DIMX - 1);
    const int iy = (int)fy0 + (DIMX - 1);
    const int ix = (int)fx0;
    if (iz < 0 || iy < 0 || ix < 0 || iz >= DIMZ || iy >= DIMY || ix >= DIMX) return;

    const long base = (long)iz * PLANE + (long)iy * DIMX + ix;
    const double wz[2] = {1.0 - fz, fz};
    const double wy[2] = {1.0 - fy, fy};
    const double wx[2] = {1.0 - fx, fx};

    double* __restrict__ wout = out + 2L * TOT;
#pragma unroll
    for (int oz = 0; oz < 2; ++oz)
#pragma unroll
        for (int oy = 0; oy < 2; ++oy)
#pragma unroll
            for (int ox = 0; ox < 2; ++ox) {
                const double dd  = wz[oz] * wy[oy] * wx[ox];
                const long   idx = base + (long)oz * PLANE + oy * DIMX + ox;
                // hardware GLOBAL_ATOMIC_ADD_F64 (non-returning, L2-resident RMW)
                unsafeAtomicAdd(out + 2 * idx,     dd * re);
                unsafeAtomicAdd(out + 2 * idx + 1, dd * im);
                unsafeAtomicAdd(wout + idx,        dd * Mw);
            }
}

// -------------------------------------------------------------------------
// Per-image constant precompute: Arot (3x2, row-major) + axx, axy, ayy.
//   Ainv = A^T * PF ; Am = Ainv[:, :2]  => Am[k][0] = PF*A[0][k], Am[k][1] = PF*A[1][k]
//   AtA  = Am^T Am  ; Arot[i][j] = Am[2-i][1-j]
// Output layout per image n (9 doubles):
//   [Arot00, Arot01, Arot10, Arot11, Arot20, Arot21, axx, axy, ayy]
// -------------------------------------------------------------------------
__global__ void bp_precompute(const double* __restrict__ A, double* __restrict__ cst) {
    int n = blockIdx.x * blockDim.x + threadIdx.x;
    if (n >= NIMG) return;
    const double* a = A + (long)n * 9;   // row-major A[n][i][j] = a[i*3+j]
    double Am[3][2];
#pragma unroll
    for (int k = 0; k < 3; ++k) {
        Am[k][0] = PFD * a[k];       // PF * A[0][k]
        Am[k][1] = PFD * a[3 + k];   // PF * A[1][k]
    }
    double axx = Am[0][0]*Am[0][0] + Am[1][0]*Am[1][0] + Am[2][0]*Am[2][0];
    double axy = Am[0][0]*Am[0][1] + Am[1][0]*Am[1][1] + Am[2][0]*Am[2][1];
    double ayy = Am[0][1]*Am[0][1] + Am[1][1]*Am[1][1] + Am[2][1]*Am[2][1];
    double* c = cst + (long)n * 9;
    c[0] = Am[2][1]; c[1] = Am[2][0];   // Arot row 0 (z)
    c[2] = Am[1][1]; c[3] = Am[1][0];   // Arot row 1 (y)
    c[4] = Am[0][1]; c[5] = Am[0][0];   // Arot row 2 (x)
    c[6] = axx; c[7] = axy; c[8] = ayy;
}

extern "C" void kernel_launch(void* const* d_in, const int* in_sizes, int n_in,
                              void* d_out, int out_size, void* d_ws, size_t ws_size,
                              hipStream_t stream) {
    (void)in_sizes; (void)n_in; (void)ws_size;
    const double* f2d_real = (const double*)d_in[0];
    const double* f2d_imag = (const double*)d_in[1];
    const double* A        = (const double*)d_in[2];
    const double* Mweight  = (const double*)d_in[3];
    double* out = (double*)d_out;
    double* cst = (double*)d_ws;   // 64*9 doubles = 4.6 KB scratch

    // Zero the accumulator volumes (harness poisons d_out with 0xAA).
    hipMemsetAsync(d_out, 0, (size_t)out_size * sizeof(double), stream);

    bp_precompute<<<1, 64, 0, stream>>>(A, cst);

    dim3 grid((HH * WW + 255) / 256, NIMG);
    bp_scatter<<<grid, 256, 0, stream>>>(f2d_real, f2d_imag, Mweight, cst, out);
}